// GPTBigCodeAttention_53266184405361
// MI455X (gfx1250) — compile-verified
//
#include <hip/hip_runtime.h>

typedef float v2f __attribute__((ext_vector_type(2)));
typedef float v8f __attribute__((ext_vector_type(8)));
typedef unsigned int v4u __attribute__((ext_vector_type(4)));
typedef int v4i __attribute__((ext_vector_type(4)));
typedef int v8i __attribute__((ext_vector_type(8)));

#define HIDDEN 2048
#define NHEADS 16
#define HDIM   128
#define SEQ    2048
#define BATCH  2
#define QKV_N  (HIDDEN + 2 * HDIM)   // 2304
#define MROWS  (BATCH * SEQ)         // 4096

__device__ __forceinline__ v8f wmma_f32(v2f a, v2f b, v8f c) {
    // D = A(16x4, f32) * B(4x16, f32) + C(16x16, f32)
    return __builtin_amdgcn_wmma_f32_16x16x4_f32(
        false, a, false, b, (short)0, c, false, false);
}

// ---- CDNA5 async global->LDS (ASYNCcnt path) --------------------------------
__device__ __forceinline__ void async_b128(void* lds_ptr, const void* gptr) {
    unsigned off = (unsigned)(unsigned long long)lds_ptr;  // addr[31:0] = LDS offset
    asm volatile("global_load_async_to_lds_b128 %0, %1, off"
                 :: "v"(off), "v"(gptr) : "memory");
}
template <int N>
__device__ __forceinline__ void wait_async() {
    asm volatile("s_wait_asynccnt %0" :: "n"(N) : "memory");
}

// ---- CDNA5 Tensor Data Mover: 2D tile (rows x 128 f32) -> LDS ---------------
// tensor_dim0 = tile_dim0 = 128 elements, row stride = QKV_N, data_size = 4B.
// LDS padding: 4 DWORDs after every 128 DWORDs -> 132-float rows (bank spread).
__device__ __forceinline__ void tdm_load_rows128(void* lds_base, const void* gaddr,
                                                 unsigned rows) {
    unsigned long long ga = (unsigned long long)gaddr;
    v4u g0;
    g0[0] = 1u;                                            // count=1 (valid user D#)
    g0[1] = (unsigned)(unsigned long long)lds_base;        // lds_addr
    g0[2] = (unsigned)(ga & 0xffffffffu);                  // global_addr[31:0]
    g0[3] = (unsigned)((ga >> 32) & 0x01ffffffu) | (2u << 30); // addr[56:32], type=2
    v8i g1;
    g1[0] = (int)((2u << 16)      // data_size = 4B
                | (1u << 20)      // pad_enable
                | (6u << 22)      // pad_interval: 128 DWORDs
                | (3u << 25));    // pad_amount: 4 DWORDs
    g1[1] = (int)(128u << 16);    // tensor_dim0 = 128 (bits 79:48 low half)
    g1[2] = (int)(rows << 16);    // tensor_dim1 = rows (bits 111:80 low half)
    g1[3] = (int)(128u << 16);    // tile_dim0 = 128 (bits 127:112)
    g1[4] = (int)rows;            // tile_dim1 = rows (bits 143:128)
    g1[5] = (int)QKV_N;           // tensor_dim0_stride[31:0]
    g1[6] = 0;                    // stride[47:32]=0, tensor_dim1_stride lo = 0
    g1[7] = 0;
    v4i z4 = {};
    v8i z8 = {};
    __builtin_amdgcn_tensor_load_to_lds(g0, g1, z4, z4, z8, 0);
}
template <int N>
__device__ __forceinline__ void wait_tensor() {
    __builtin_amdgcn_s_wait_tensorcnt((short)N);
    asm volatile("" ::: "memory");
}

// ---------------------------------------------------------------------------
// C = A(MxK) * B(KxN) + bias. Block tile 256x128, BK=16, 8 waves of 64x64.
// A/B tiles staged to LDS with async b128 copies, double buffered.
// ---------------------------------------------------------------------------
__global__ __launch_bounds__(256) void gemm_bias_async(
    const float* __restrict__ A, const float* __restrict__ B,
    const float* __restrict__ bias, float* __restrict__ C,
    int M, int N, int K)
{
    __shared__ float sA[2][256][20];   // 20-float rows: 16B aligned, bank-spread
    __shared__ float sB[2][16][128];

    const int t    = threadIdx.x;
    const int lane = t & 31;
    const int wave = t >> 5;
    const int l15  = lane & 15;
    const int grp  = lane >> 4;
    const int kof  = grp << 1;                 // 0 or 2
    const int wm   = wave & 3;                 // 4 row-waves (64 rows each)
    const int wn   = wave >> 2;                // 2 col-waves (64 cols each)
    const int m0   = blockIdx.y * 256;
    const int n0   = blockIdx.x * 128;

    const int ar = t >> 2, ac = (t & 3) * 4;   // A: 4 rounds of 64 rows x 16 cols

    auto stage = [&](int buf, int k0) {
        #pragma unroll
        for (int r = 0; r < 4; r++)
            async_b128(&sA[buf][r * 64 + ar][ac],
                       A + (size_t)(m0 + r * 64 + ar) * K + k0 + ac);
        #pragma unroll
        for (int r = 0; r < 2; r++) {
            const int idx = r * 256 + t;
            const int brow = idx >> 5, bcol = (idx & 31) * 4;
            async_b128(&sB[buf][brow][bcol],
                       B + (size_t)(k0 + brow) * N + n0 + bcol);
        }
    };

    v8f acc[4][4] = {};
    stage(0, 0);
    const int nk = K >> 4;
    for (int kb = 0; kb < nk; kb++) {
        const int buf = kb & 1;
        if (kb + 1 < nk) { stage(buf ^ 1, (kb + 1) << 4); wait_async<6>(); }
        else             { wait_async<0>(); }
        __syncthreads();
        #pragma unroll
        for (int c = 0; c < 4; c++) {
            const int kk = c * 4 + kof;
            v2f a[4];
            #pragma unroll
            for (int i = 0; i < 4; i++)
                a[i] = *(const v2f*)&sA[buf][wm * 64 + i * 16 + l15][kk];
            v2f bf[4];
            #pragma unroll
            for (int j = 0; j < 4; j++) {
                bf[j].x = sB[buf][kk    ][wn * 64 + j * 16 + l15];
                bf[j].y = sB[buf][kk + 1][wn * 64 + j * 16 + l15];
            }
            #pragma unroll
            for (int i = 0; i < 4; i++)
                #pragma unroll
                for (int j = 0; j < 4; j++)
                    acc[i][j] = wmma_f32(a[i], bf[j], acc[i][j]);
        }
        __syncthreads();                        // buf consumed by all waves
    }

    #pragma unroll
    for (int ti = 0; ti < 4; ti++) {
        #pragma unroll
        for (int tj = 0; tj < 4; tj++) {
            const int col = n0 + wn * 64 + tj * 16 + l15;
            const float bv = bias[col];
            const int r0 = m0 + wm * 64 + ti * 16 + grp * 8;
            #pragma unroll
            for (int j = 0; j < 8; j++)
                C[(size_t)(r0 + j) * N + col] = acc[ti][tj][j] + bv;
        }
    }
}

// ---------------------------------------------------------------------------
// Fused multi-query causal attention, flash style. One wave per
// (batch, head, 16-query tile). K/V 16x128 blocks pulled by the Tensor Data
// Mover (1 TDM descriptor each), double buffered on TENSORcnt.
// ---------------------------------------------------------------------------
__global__ __launch_bounds__(32) void mqa_attn(
    const float* __restrict__ qkv, float* __restrict__ out)
{
    __shared__ float sK[2][16][132];   // TDM-padded rows (4 DW pad per 128 DW)
    __shared__ float sV[2][16][132];
    __shared__ float pls[16][17];      // P tile staging

    const int lane = threadIdx.x & 31;
    const int l15  = lane & 15;
    const int grp  = lane >> 4;
    const int kof  = grp << 1;

    const int qt = blockIdx.x & 127;
    const int h  = (blockIdx.x >> 7) & 15;
    const int b  = blockIdx.x >> 11;
    const int q0 = qt << 4;

    const float* Qb = qkv + (size_t)(b * SEQ) * QKV_N;
    const float* Kb = Qb + HIDDEN;
    const float* Vb = Qb + HIDDEN + HDIM;

    // Q tile resident as 32 A-fragments (K-dim = hd = 128)
    v2f qf[32];
    {
        const float* qrow = Qb + (size_t)(q0 + l15) * QKV_N + h * HDIM + kof;
        #pragma unroll
        for (int c = 0; c < 32; c++)
            qf[c] = *(const v2f*)(qrow + c * 4);
    }

    auto stageKV = [&](int buf, int k0) {
        tdm_load_rows128(&sK[buf][0][0], Kb + (size_t)k0 * QKV_N, 16u);
        tdm_load_rows128(&sV[buf][0][0], Vb + (size_t)k0 * QKV_N, 16u);
    };

    v8f o[8];
    float m[8], l[8];
    #pragma unroll
    for (int ti = 0; ti < 8; ti++) o[ti] = (v8f){};
    #pragma unroll
    for (int j = 0; j < 8; j++) { m[j] = -1e30f; l[j] = 0.0f; }

    const float scale = 0.08838834764831845f;   // 1/sqrt(128)

    stageKV(0, 0);
    for (int kb = 0; kb <= qt; kb++) {
        const int buf = kb & 1;
        const int k0  = kb << 4;
        if (kb < qt) { stageKV(buf ^ 1, (kb + 1) << 4); wait_tensor<2>(); }
        else         { wait_tensor<0>(); }

        __syncthreads();                        // pls WAR (single wave)

        // ---- scores: S = Q (16x128) * K_blk^T (128x16), from LDS ----
        v8f s = {};
        #pragma unroll
        for (int c = 0; c < 32; c++) {
            v2f bf = *(const v2f*)&sK[buf][l15][c * 4 + kof];
            s = wmma_f32(qf[c], bf, s);
        }

        // ---- scale + causal mask (diagonal block only) ----
        const int colk  = k0 + l15;
        const int rbase = q0 + grp * 8;
        #pragma unroll
        for (int j = 0; j < 8; j++) {
            float v = s[j] * scale;
            if (kb == qt && colk > rbase + j) v = -1e30f;
            s[j] = v;
        }

        // ---- online softmax (row stats across each 16-lane half) ----
        float corr[8];
        #pragma unroll
        for (int j = 0; j < 8; j++) {
            float rmax = s[j];
            rmax = fmaxf(rmax, __shfl_xor(rmax, 1, 32));
            rmax = fmaxf(rmax, __shfl_xor(rmax, 2, 32));
            rmax = fmaxf(rmax, __shfl_xor(rmax, 4, 32));
            rmax = fmaxf(rmax, __shfl_xor(rmax, 8, 32));
            const float mnew = fmaxf(m[j], rmax);
            const float p    = __expf(s[j] - mnew);
            const float cj   = __expf(m[j] - mnew);
            float rsum = p;
            rsum += __shfl_xor(rsum, 1, 32);
            rsum += __shfl_xor(rsum, 2, 32);
            rsum += __shfl_xor(rsum, 4, 32);
            rsum += __shfl_xor(rsum, 8, 32);
            l[j] = l[j] * cj + rsum;
            m[j] = mnew;
            corr[j] = cj;
            pls[grp * 8 + j][l15] = p;          // P[query_row][key_col]
        }
        #pragma unroll
        for (int ti = 0; ti < 8; ti++)
            #pragma unroll
            for (int j = 0; j < 8; j++)
                o[ti][j] *= corr[j];

        __syncthreads();

        // ---- re-fragment P into A-layout ----
        v2f pa[4];
        #pragma unroll
        for (int c = 0; c < 4; c++) {
            const int kk = c * 4 + kof;
            pa[c].x = pls[l15][kk];
            pa[c].y = pls[l15][kk + 1];
        }

        // ---- O += P (16x16) * V_blk (16x128), V from LDS ----
        #pragma unroll
        for (int ti = 0; ti < 8; ti++) {
            #pragma unroll
            for (int c = 0; c < 4; c++) {
                const int kr = c * 4 + kof;
                v2f bv;
                bv.x = sV[buf][kr    ][ti * 16 + l15];
                bv.y = sV[buf][kr + 1][ti * 16 + l15];
                o[ti] = wmma_f32(pa[c], bv, o[ti]);
            }
        }
    }

    // ---- normalize and store attn_output[b, q, h*128 + d] ----
    #pragma unroll
    for (int j = 0; j < 8; j++) l[j] = 1.0f / l[j];
    const int orow = b * SEQ + q0 + grp * 8;
    #pragma unroll
    for (int ti = 0; ti < 8; ti++)
        #pragma unroll
        for (int j = 0; j < 8; j++)
            out[(size_t)(orow + j) * HIDDEN + h * HDIM + ti * 16 + l15] = o[ti][j] * l[j];
}

// ---------------------------------------------------------------------------
extern "C" void kernel_launch(void* const* d_in, const int* in_sizes, int n_in,
                              void* d_out, int out_size, void* d_ws, size_t ws_size,
                              hipStream_t stream) {
    const float* hidden   = (const float*)d_in[0];
    // d_in[1] = attention_mask (causal) — computed analytically, not read.
    const float* c_attn_w = (const float*)d_in[2];
    const float* c_attn_b = (const float*)d_in[3];
    const float* c_proj_w = (const float*)d_in[4];
    const float* c_proj_b = (const float*)d_in[5];
    float* outp = (float*)d_out;

    float* qkv  = (float*)d_ws;                         // 4096 x 2304 f32
    float* attn = qkv + (size_t)MROWS * QKV_N;          // 4096 x 2048 f32

    // 1) QKV projection: (4096 x 2048) @ (2048 x 2304) + bias
    {
        dim3 grid(QKV_N / 128, MROWS / 256);            // 18 x 16
        gemm_bias_async<<<grid, 256, 0, stream>>>(
            hidden, c_attn_w, c_attn_b, qkv, MROWS, QKV_N, HIDDEN);
    }
    // 2) fused causal multi-query attention (TDM-fed)
    mqa_attn<<<BATCH * NHEADS * (SEQ / 16), 32, 0, stream>>>(qkv, attn);
    // 3) output projection: (4096 x 2048) @ (2048 x 2048) + bias
    {
        dim3 grid(HIDDEN / 128, MROWS / 256);           // 16 x 16
        gemm_bias_async<<<grid, 256, 0, stream>>>(
            attn, c_proj_w, c_proj_b, outp, MROWS, HIDDEN, HIDDEN);
    }
}